// EpisodicSlotReader_63823214018884
// MI455X (gfx1250) — compile-verified
//
#include <hip/hip_runtime.h>
#include <math.h>

#define B_ 32
#define S_ 8192
#define D_ 512
#define NCHUNK 64
#define SLOTS_PER_CHUNK (S_ / NCHUNK) // 128

typedef float v2f __attribute__((ext_vector_type(2)));
typedef float v4f __attribute__((ext_vector_type(4)));
typedef float v8f __attribute__((ext_vector_type(8)));

// ---------------- Kernel 0: per-batch inverse query norm ----------------
__global__ void qnorm_kernel(const float* __restrict__ q, float* __restrict__ inv_qn) {
    __shared__ float red[256];
    const int b = blockIdx.x, tid = threadIdx.x;
    const float* qb = q + (size_t)b * D_;
    float ss = 0.f;
    for (int i = tid; i < D_; i += 256) { float v = qb[i]; ss = fmaf(v, v, ss); }
    red[tid] = ss; __syncthreads();
    for (int off = 128; off > 0; off >>= 1) {
        if (tid < off) red[tid] += red[tid + off];
        __syncthreads();
    }
    if (tid == 0) inv_qn[b] = 1.f / (sqrtf(red[0]) + 1e-6f);
}

// ---------------- Kernel 1: logits via f32 WMMA over streamed keys ----------------
// One wave handles 16 slots. B operand = key tile (4x16), A operand = q chunk
// replicated across all 16 rows -> every row of C holds the 16 slot dots.
// Lane l (kh = l>>4, srow = l&15) loads keys[slot srow][8t + 4*kh .. +3] as b128,
// which feeds two chained 16x16x4 WMMAs per iteration. Sum-of-squares of the
// same registers gives ||k|| after a shfl_xor(16) combine.
__global__ void logits_kernel(const float* __restrict__ q,
                              const float* __restrict__ keys,
                              const float* __restrict__ age,
                              const float* __restrict__ strength,
                              const float* __restrict__ inv_qn,
                              float* __restrict__ logits) {
    __shared__ float lq[D_];
    const int b = blockIdx.y;
    const int tid = threadIdx.x; // 256 threads = 8 waves
    // Stage q[b] into LDS (256 threads x float2)
    ((float2*)lq)[tid] = ((const float2*)(q + (size_t)b * D_))[tid];
    __syncthreads();

    const int wave = tid >> 5;
    const int lane = tid & 31;
    const int kh   = lane >> 4;   // which K-half of the WMMA operands
    const int srow = lane & 15;   // slot within the 16-slot group
    const int sbase = (blockIdx.x * 8 + wave) * 16;
    const int slot  = sbase + srow;

    const float* kp = keys + ((size_t)b * S_ + slot) * D_ + kh * 4;
    const float* qp = lq + kh * 4;

    v8f c = {0.f, 0.f, 0.f, 0.f, 0.f, 0.f, 0.f, 0.f};
    float ss = 0.f;
#pragma unroll 4
    for (int t = 0; t < D_ / 8; ++t) {
        v4f kv = __builtin_nontemporal_load((const v4f*)(kp + 8 * t));
        v4f qv = *(const v4f*)(qp + 8 * t); // ds_load_b128
        v2f a1 = {qv.x, qv.y}, b1 = {kv.x, kv.y};
        c = __builtin_amdgcn_wmma_f32_16x16x4_f32(false, a1, false, b1,
                                                  (short)0, c, false, false);
        v2f a2 = {qv.z, qv.w}, b2 = {kv.z, kv.w};
        c = __builtin_amdgcn_wmma_f32_16x16x4_f32(false, a2, false, b2,
                                                  (short)0, c, false, false);
        ss = fmaf(kv.x, kv.x, ss);
        ss = fmaf(kv.y, kv.y, ss);
        ss = fmaf(kv.z, kv.z, ss);
        ss = fmaf(kv.w, kv.w, ss);
    }
    const float dot = c[0];                 // every lane: dot for slot (lane&15)
    ss += __shfl_xor(ss, 16, 32);           // combine the two K-halves

    if (lane < 16) {
        const size_t idx = (size_t)b * S_ + slot;
        const float sim = dot * inv_qn[b] / (sqrtf(ss) + 1e-6f);
        const float st  = strength[idx];
        const float ag  = age[idx];
        const float stc = fminf(fmaxf(st, 1e-3f), 1e9f);
        float lg = sim + 0.5f * logf(stc) - 0.02f * ag;
        if (!(st > 1e-3f)) lg -= 1000.f;
        logits[idx] = lg;
    }
}

// ---------------- Kernel 2: softmax over S per batch ----------------
__global__ void softmax_kernel(const float* __restrict__ logits, float* __restrict__ w) {
    __shared__ float red[256];
    const int b = blockIdx.x, tid = threadIdx.x;
    const float* lb = logits + (size_t)b * S_;
    float m = -INFINITY;
    float ev[S_ / 256];
    for (int i = tid; i < S_; i += 256) m = fmaxf(m, lb[i]);
    red[tid] = m; __syncthreads();
    for (int off = 128; off > 0; off >>= 1) {
        if (tid < off) red[tid] = fmaxf(red[tid], red[tid + off]);
        __syncthreads();
    }
    m = red[0]; __syncthreads();
    float s = 0.f;
    int j = 0;
    for (int i = tid; i < S_; i += 256, ++j) {
        const float e = expf(lb[i] - m);
        ev[j] = e;
        s += e;
    }
    red[tid] = s; __syncthreads();
    for (int off = 128; off > 0; off >>= 1) {
        if (tid < off) red[tid] += red[tid + off];
        __syncthreads();
    }
    const float inv = 1.f / red[0];
    j = 0;
    for (int i = tid; i < S_; i += 256, ++j) w[(size_t)b * S_ + i] = ev[j] * inv;
}

// ---------------- Kernel 3: partial weighted read over an S chunk ----------------
__global__ void readpart_kernel(const float* __restrict__ vals,
                                const float* __restrict__ w,
                                float* __restrict__ part) {
    __shared__ float lw[SLOTS_PER_CHUNK];
    const int b = blockIdx.y, chunk = blockIdx.x, tid = threadIdx.x; // 128 threads
    const int s0 = chunk * SLOTS_PER_CHUNK;
    lw[tid] = w[(size_t)b * S_ + s0 + tid];
    __syncthreads();
    const int d0 = tid * 4;
    const float* vp = vals + ((size_t)b * S_ + s0) * D_ + d0;
    v4f acc = {0.f, 0.f, 0.f, 0.f};
    for (int s = 0; s < SLOTS_PER_CHUNK; ++s) {
        const v4f v = __builtin_nontemporal_load((const v4f*)(vp + (size_t)s * D_));
        const float wv = lw[s];
        acc.x = fmaf(wv, v.x, acc.x);
        acc.y = fmaf(wv, v.y, acc.y);
        acc.z = fmaf(wv, v.z, acc.z);
        acc.w = fmaf(wv, v.w, acc.w);
    }
    *(v4f*)(part + ((size_t)chunk * B_ + b) * D_ + d0) = acc;
}

// ---------------- Kernel 4: chunk reduce + RMSNorm * scale ----------------
__global__ void finalize_kernel(const float* __restrict__ part,
                                const float* __restrict__ scale,
                                float* __restrict__ outr) {
    __shared__ float red[128];
    const int b = blockIdx.x, tid = threadIdx.x; // 128 threads
    const int d0 = tid * 4;
    v4f acc = {0.f, 0.f, 0.f, 0.f};
    for (int c = 0; c < NCHUNK; ++c) {
        const v4f p = *(const v4f*)(part + ((size_t)c * B_ + b) * D_ + d0);
        acc += p;
    }
    red[tid] = acc.x * acc.x + acc.y * acc.y + acc.z * acc.z + acc.w * acc.w;
    __syncthreads();
    for (int off = 64; off > 0; off >>= 1) {
        if (tid < off) red[tid] += red[tid + off];
        __syncthreads();
    }
    const float inv = 1.f / sqrtf(red[0] / (float)D_ + 1e-6f);
    const v4f sc = *(const v4f*)(scale + d0);
    v4f o = acc * inv * sc;
    *(v4f*)(outr + (size_t)b * D_ + d0) = o;
}

extern "C" void kernel_launch(void* const* d_in, const int* in_sizes, int n_in,
                              void* d_out, int out_size, void* d_ws, size_t ws_size,
                              hipStream_t stream) {
    (void)in_sizes; (void)n_in; (void)out_size; (void)ws_size;
    const float* q        = (const float*)d_in[0];
    const float* keys     = (const float*)d_in[1];
    const float* vals     = (const float*)d_in[2];
    const float* age      = (const float*)d_in[3];
    const float* strength = (const float*)d_in[4];
    const float* scale    = (const float*)d_in[5];

    float* out        = (float*)d_out;
    float* out_read   = out;                     // [B, D]
    float* out_w      = out + B_ * D_;           // [B, S]
    float* out_logits = out + B_ * D_ + B_ * S_; // [B, S]

    float* ws     = (float*)d_ws;
    float* inv_qn = ws;        // 32 floats (64 reserved for alignment)
    float* part   = ws + 64;   // [NCHUNK, B, D] = 4 MB

    qnorm_kernel<<<B_, 256, 0, stream>>>(q, inv_qn);

    dim3 g1(S_ / 128, B_); // 64 x 32 blocks, 8 waves each
    logits_kernel<<<g1, 256, 0, stream>>>(q, keys, age, strength, inv_qn, out_logits);

    softmax_kernel<<<B_, 256, 0, stream>>>(out_logits, out_w);

    dim3 g3(NCHUNK, B_); // 64 x 32 blocks
    readpart_kernel<<<g3, 128, 0, stream>>>(vals, out_w, part);

    finalize_kernel<<<B_, 128, 0, stream>>>(part, scale, out_read);
}